// HarmonicSynth_34007551050062
// MI455X (gfx1250) — compile-verified
//
#include <hip/hip_runtime.h>
#include <math.h>

typedef __attribute__((ext_vector_type(2))) float v2f;
typedef __attribute__((ext_vector_type(8))) float v8f;

#define NBATCH 16
#define NH 64
#define TIN 400
#define UPF 160
#define TOUT 64000
#define SRATE 16000.0f
#define TWO_PI 6.28318530717958647692f

// ---------------------------------------------------------------------------
// Kernel 1: per-batch exclusive prefix of segment sums of the upsampled pitch.
// Regions: r=0 -> t in [0,80) (clamped, 80*p[0]); r=1..399 -> 160 samples,
// sum = 80*(p[r-1]+p[r]); r=400 -> t in [63920,64000), 80*p[399].
// Pre[b][r] = sum of pitch_up over all samples before region r's start.
// One wave per batch, shfl-based scan over 401 entries.
// ---------------------------------------------------------------------------
__global__ __launch_bounds__(512) void hs_prefix(const float* __restrict__ pitch,
                                                 float* __restrict__ pre) {
    const int bat  = threadIdx.x >> 5;   // 16 waves, one per batch
    const int lane = threadIdx.x & 31;
    const float* p = pitch + bat * TIN;
    float* P = pre + bat * 401;

    float base = 0.0f;
    for (int g = 0; g < 416; g += 32) {
        int idx = g + lane;
        float v = 0.0f;
        if (idx == 0)        v = 80.0f * p[0];
        else if (idx < 400)  v = 80.0f * (p[idx - 1] + p[idx]);
        // idx >= 400: segment sum not needed for any Pre entry -> 0
        // inclusive scan within wave
        float inc = v;
        #pragma unroll
        for (int d = 1; d < 32; d <<= 1) {
            float t = __shfl_up(inc, d, 32);
            if (lane >= d) inc += t;
        }
        float excl = inc - v;
        if (idx <= 400) P[idx] = base + excl;
        base += __shfl(inc, 31, 32);
    }
}

// ---------------------------------------------------------------------------
// Kernel 2: one wave per (batch, 16-sample tile). Each lane: sample m=lane&15,
// two harmonic streams (lanes<16: h%4==1,2 ; lanes>=16: h%4==3,0), rotation
// recurrence by 4*phi, accumulate masked numerator/denominator partials
// (numerator pre-scaled by per-sample amplitude), then
// V_WMMA_F32_16X16X4_F32 (B = ones) reduces K=4 and transposes 16 samples.
// Epilogue: uniform rcp-divide, lanes 0 and 16 store 8 floats each (2x b128).
// ---------------------------------------------------------------------------
__global__ __launch_bounds__(256) void hs_main(const float* __restrict__ amp,
                                               const float* __restrict__ alphas,
                                               const float* __restrict__ pitch,
                                               const float* __restrict__ pre,
                                               float* __restrict__ out) {
    const int b    = blockIdx.y;
    const int lane = threadIdx.x & 31;
    const int tile = blockIdx.x * 8 + (threadIdx.x >> 5);   // 0..3999
    const int t0   = tile * 16;
    const int m    = lane & 15;
    const int t    = t0 + m;

    // --- region decode (uniform within a tile: boundaries are 16-aligned) ---
    int r, j, i0;
    if (t < 80)              { r = 0;                   j = t;                      i0 = 0;     }
    else if (t < 63920)      { r = 1 + (t - 80) / UPF;  j = (t - 80) - (r-1)*UPF;   i0 = r - 1; }
    else                     { r = 400;                 j = t - 63920;              i0 = 399;   }
    const int  i1       = min(i0 + 1, TIN - 1);
    const bool interior = (r >= 1) && (r <= 399);
    const float w  = interior ? ((float)j + 0.5f) * (1.0f / UPF) : 0.0f;
    const float wc = 1.0f - w;

    // --- phase via closed-form piecewise-quadratic cumsum ---
    const float* pb = pitch + b * TIN;
    const float p0 = pb[i0], p1 = pb[i1];
    const float d  = interior ? (p1 - p0) : 0.0f;
    const float jp1 = (float)(j + 1);
    const float partial = jp1 * p0 + d * (jp1 * jp1) * (1.0f / 320.0f);
    const float S   = pre[b * 401 + r] + partial;
    const float phi = S * (TWO_PI / SRATE);

    const float pitch_t = fmaf(w, d, p0);
    const float pn      = pitch_t * (1.0f / SRATE);   // normalized frequency

    const float* ab = amp + b * TIN;
    const float amp_t = ab[i0] * wc + ab[i1] * w;

    // --- harmonic streams: lane half chooses K-slot pair per f32 A layout ---
    const int o0 = (lane < 16) ? 0 : 2;               // K = o0 (VGPR0), o0+1 (VGPR1)
    float sA, cA, sB, cB, s4, c4;
    __sincosf((float)(o0 + 1) * phi, &sA, &cA);       // h0 = 1 or 3
    __sincosf((float)(o0 + 2) * phi, &sB, &cB);       // h0 = 2 or 4
    __sincosf(4.0f * phi, &s4, &c4);                  // rotation step (dh = 4)

    float nAcc0 = 0.0f, nAcc1 = 0.0f, dAcc0 = 0.0f, dAcc1 = 0.0f;
    const float* rowA = alphas + ((size_t)(b * NH + o0)) * TIN;
    float hAf = (float)(1 + o0);

    #pragma unroll 4
    for (int c = 0; c < 16; ++c) {
        const float* rowB = rowA + TIN;
        float aA = rowA[i0] * wc + rowA[i1] * w;      // upsampled alpha, stream A
        float aB = rowB[i0] * wc + rowB[i1] * w;      // stream B
        aA = (pn * hAf          < 0.5f) ? aA : 0.0f;  // aliasing mask
        aB = (pn * (hAf + 1.0f) < 0.5f) ? aB : 0.0f;
        dAcc0 += aA;                 dAcc1 += aB;
        nAcc0 = fmaf(sA, aA, nAcc0); nAcc1 = fmaf(sB, aB, nAcc1);
        // rotate both streams by 4*phi
        float sAn = fmaf(sA, c4,  cA * s4);
        float cAn = fmaf(cA, c4, -sA * s4);
        float sBn = fmaf(sB, c4,  cB * s4);
        float cBn = fmaf(cB, c4, -sB * s4);
        sA = sAn; cA = cAn; sB = sBn; cB = cBn;
        hAf  += 4.0f;
        rowA += 4 * TIN;
    }

    // Fold amplitude into the numerator: row m of A is owned by lanes m and
    // m+16, both of which computed amp_t for this same m.
    nAcc0 *= amp_t;
    nAcc1 *= amp_t;

    // --- K=4 reduce + 16-sample transpose via f32 WMMA, B = ones ---
    v2f anum = {nAcc0, nAcc1};
    v2f aden = {dAcc0, dAcc1};
    v2f ones = {1.0f, 1.0f};
    v8f zero = {};
    v8f Dn = __builtin_amdgcn_wmma_f32_16x16x4_f32(false, anum, false, ones,
                                                   (short)0, zero, false, false);
    v8f Dd = __builtin_amdgcn_wmma_f32_16x16x4_f32(false, aden, false, ones,
                                                   (short)0, zero, false, false);

    // D layout: VGPR rr -> M=rr (lanes 0-15) / M=8+rr (lanes 16-31); all N
    // columns identical (B=ones). Uniform divide (no divergence, static
    // indices), then lane 0 stores samples 0..7 and lane 16 stores 8..15.
    float y0 = Dn[0] * __builtin_amdgcn_rcpf(Dd[0]);
    float y1 = Dn[1] * __builtin_amdgcn_rcpf(Dd[1]);
    float y2 = Dn[2] * __builtin_amdgcn_rcpf(Dd[2]);
    float y3 = Dn[3] * __builtin_amdgcn_rcpf(Dd[3]);
    float y4 = Dn[4] * __builtin_amdgcn_rcpf(Dd[4]);
    float y5 = Dn[5] * __builtin_amdgcn_rcpf(Dd[5]);
    float y6 = Dn[6] * __builtin_amdgcn_rcpf(Dd[6]);
    float y7 = Dn[7] * __builtin_amdgcn_rcpf(Dd[7]);

    if ((lane & 15) == 0) {
        float* dst = out + (size_t)b * TOUT + t0 + ((lane >> 4) << 3);
        float4 lo = {y0, y1, y2, y3};
        float4 hi = {y4, y5, y6, y7};
        *(float4*)(dst)     = lo;   // 16B-aligned: t0 is a multiple of 16
        *(float4*)(dst + 4) = hi;
    }
}

extern "C" void kernel_launch(void* const* d_in, const int* in_sizes, int n_in,
                              void* d_out, int out_size, void* d_ws, size_t ws_size,
                              hipStream_t stream) {
    (void)in_sizes; (void)n_in; (void)out_size; (void)ws_size;
    const float* amp    = (const float*)d_in[0];   // [16,1,400]
    const float* alphas = (const float*)d_in[1];   // [16,64,400]
    const float* pitch  = (const float*)d_in[2];   // [16,1,400]
    float* out = (float*)d_out;                    // [16,1,64000]
    float* pre = (float*)d_ws;                     // 16*401 floats

    hs_prefix<<<1, 512, 0, stream>>>(pitch, pre);
    hs_main<<<dim3(500, 16, 1), 256, 0, stream>>>(amp, alphas, pitch, pre, out);
}